// MDE_38663295599392
// MI455X (gfx1250) — compile-verified
//
#include <hip/hip_runtime.h>
#include <stdint.h>

// ---------------------------------------------------------------------------
// MDE quadratic distortion: mean_k w_k * ||X[i_k] - X[j_k]||^2
// Memory floor ~9.3us (216MB @ 23.3TB/s), but per-edge instruction count is
// the practical limit -> minimize instructions/edge.
// CDNA5 paths: global_load_async_to_lds_b128 (ASYNCcnt pipeline, GVS
// saddr form, shared INST_OFFSET feeds both LDS and global sides) with NT
// hints so the 200MB edge/weight stream never evicts the 16MB X table from
// the 192MB L2. Gathers use default (RT) b128 loads. Deterministic 2-stage
// reduction (no float atomics).
// ---------------------------------------------------------------------------

#define BLOCK   256
#define WPB     8            // waves per block (wave32)
#define TILE    128          // edges per wave per tile (4 per lane)
#define NB_MAX  1024         // blocks for the partial kernel

__global__ __launch_bounds__(BLOCK)
void mde_partial_kernel(const float* __restrict__ X,
                        const unsigned long long* __restrict__ edges, // [n,2] i64
                        const float* __restrict__ weights,
                        float* __restrict__ partials,
                        long long n)
{
    // Per-wave double buffers: edges 2*2048B, weights 2*512B -> 5KB/wave, 40KB/block.
    __shared__ __align__(16) unsigned char ebuf[WPB][2][TILE * 16];
    __shared__ __align__(16) unsigned char wbuf[WPB][2][TILE * 4];
    __shared__ float wave_sums[WPB];

    const int       lane   = threadIdx.x & 31;
    const int       wave   = threadIdx.x >> 5;
    const long long nwaves = (long long)gridDim.x * WPB;
    const long long gw     = (long long)blockIdx.x * WPB + wave;   // global wave id
    const long long full   = n / TILE;                              // full tiles

    const unsigned long long edges_base   = (unsigned long long)(const void*)edges;
    const unsigned long long weights_base = (unsigned long long)(const void*)weights;

    // Issue one 128-edge tile: 5 async instructions, lane-parallel.
    //   edge sub-load k: lane l copies edge (t*128 + k*32 + l), 16B each.
    //     INST_OFFSET (k*512) advances global AND LDS addresses together.
    //   weight load: lane l copies weights[t*128 + 4l .. +3] (16B).
    auto issue_tile = [&](long long t, int b) {
        unsigned lds_e  = (unsigned)(size_t)&ebuf[wave][b][lane * 16];
        unsigned voff_e = (unsigned)(t * TILE + lane) * 16u;
        unsigned lds_w  = (unsigned)(size_t)&wbuf[wave][b][lane * 16];
        unsigned voff_w = (unsigned)(t * TILE + lane * 4) * 4u;
        asm volatile("global_load_async_to_lds_b128 %0, %1, %2 offset:0 th:TH_LOAD_NT"
                     :: "v"(lds_e), "v"(voff_e), "s"(edges_base) : "memory");
        asm volatile("global_load_async_to_lds_b128 %0, %1, %2 offset:512 th:TH_LOAD_NT"
                     :: "v"(lds_e), "v"(voff_e), "s"(edges_base) : "memory");
        asm volatile("global_load_async_to_lds_b128 %0, %1, %2 offset:1024 th:TH_LOAD_NT"
                     :: "v"(lds_e), "v"(voff_e), "s"(edges_base) : "memory");
        asm volatile("global_load_async_to_lds_b128 %0, %1, %2 offset:1536 th:TH_LOAD_NT"
                     :: "v"(lds_e), "v"(voff_e), "s"(edges_base) : "memory");
        asm volatile("global_load_async_to_lds_b128 %0, %1, %2 offset:0 th:TH_LOAD_NT"
                     :: "v"(lds_w), "v"(voff_w), "s"(weights_base) : "memory");
    };

    float acc = 0.0f;

    // Consume one tile from LDS: conflict-free strided reads, immediate offsets.
    auto consume_tile = [&](int b) {
        const unsigned char* eb = &ebuf[wave][b][0];
        const unsigned char* wb = &wbuf[wave][b][0];
#pragma unroll
        for (int k = 0; k < 4; ++k) {
            ulonglong2 e = *(const ulonglong2*)(eb + (k * 32 + lane) * 16); // ds_load_b128
            float      w = *(const float*)(wb + (k * 32 + lane) * 4);       // ds_load_b32
            unsigned i = (unsigned)e.x;
            unsigned j = (unsigned)e.y;
            float4 xi = *(const float4*)(X + (size_t)i * 4);  // L2-resident gathers
            float4 xj = *(const float4*)(X + (size_t)j * 4);
            float dx = xi.x - xj.x, dy = xi.y - xj.y;
            float dz = xi.z - xj.z, dw = xi.w - xj.w;
            float d2 = dx*dx + dy*dy + dz*dz + dw*dw;
            acc += w * d2;
        }
    };

    // Double-buffered pipeline, hand-unrolled 2x so buffer indices are constant.
    long long t = gw;
    if (t < full) issue_tile(t, 0);
    while (t < full) {
        long long tn = t + nwaves;
        if (tn < full) {
            issue_tile(tn, 1);
            asm volatile("s_wait_asynccnt 0x5" ::: "memory");
        } else {
            asm volatile("s_wait_asynccnt 0x0" ::: "memory");
        }
        consume_tile(0);
        if (tn >= full) break;
        long long tnn = tn + nwaves;
        if (tnn < full) {
            issue_tile(tnn, 0);
            asm volatile("s_wait_asynccnt 0x5" ::: "memory");
        } else {
            asm volatile("s_wait_asynccnt 0x0" ::: "memory");
        }
        consume_tile(1);
        t = tnn;
    }

    // Tail (< 128 edges): handled by global wave 0, direct loads, masked.
    if (gw == 0) {
        for (long long e2 = full * TILE + lane; e2 < n; e2 += 32) {
            ulonglong2 ep = *(const ulonglong2*)(edges + e2 * 2);
            float      w  = weights[e2];
            unsigned i = (unsigned)ep.x;
            unsigned j = (unsigned)ep.y;
            float4 xi = *(const float4*)(X + (size_t)i * 4);
            float4 xj = *(const float4*)(X + (size_t)j * 4);
            float dx = xi.x - xj.x, dy = xi.y - xj.y;
            float dz = xi.z - xj.z, dw = xi.w - xj.w;
            acc += w * (dx*dx + dy*dy + dz*dz + dw*dw);
        }
    }

    // Deterministic intra-wave reduction (wave32).
    for (int m = 16; m >= 1; m >>= 1)
        acc += __shfl_xor(acc, m, 32);
    if (lane == 0) wave_sums[wave] = acc;
    __syncthreads();
    if (threadIdx.x == 0) {
        float s = 0.0f;
        for (int wv = 0; wv < WPB; ++wv) s += wave_sums[wv];
        partials[blockIdx.x] = s;
    }
}

__global__ __launch_bounds__(256)
void mde_final_kernel(const float* __restrict__ partials, int nb,
                      float inv_n, float* __restrict__ out)
{
    __shared__ float s[256];
    float a = 0.0f;
    for (int idx = threadIdx.x; idx < nb; idx += 256) a += partials[idx];
    s[threadIdx.x] = a;
    __syncthreads();
    for (int k = 128; k > 0; k >>= 1) {
        if (threadIdx.x < k) s[threadIdx.x] += s[threadIdx.x + k];
        __syncthreads();
    }
    if (threadIdx.x == 0) out[0] = s[0] * inv_n;
}

extern "C" void kernel_launch(void* const* d_in, const int* in_sizes, int n_in,
                              void* d_out, int out_size, void* d_ws, size_t ws_size,
                              hipStream_t stream)
{
    const float*              X       = (const float*)d_in[0];
    const unsigned long long* edges   = (const unsigned long long*)d_in[1]; // int64 pairs
    const float*              weights = (const float*)d_in[2];
    const long long n = (long long)in_sizes[2];   // number of edges

    float* partials = (float*)d_ws;
    int nb = NB_MAX;
    size_t cap = ws_size / sizeof(float);
    if ((size_t)nb > cap) nb = (int)cap;
    if (nb < 1) nb = 1;

    mde_partial_kernel<<<nb, BLOCK, 0, stream>>>(X, edges, weights, partials, n);
    mde_final_kernel<<<1, 256, 0, stream>>>(partials, nb,
                                            1.0f / (float)n, (float*)d_out);
}